// GCN_32366873542797
// MI455X (gfx1250) — compile-verified
//
#include <hip/hip_runtime.h>
#include <math.h>

typedef int   v2i __attribute__((ext_vector_type(2)));
typedef float v2f __attribute__((ext_vector_type(2)));
typedef float v4f __attribute__((ext_vector_type(4)));
typedef float v8f __attribute__((ext_vector_type(8)));

#define NF 256

// ---------------- utility kernels ----------------
__global__ void fill_f32(float* __restrict__ p, float v, int n) {
  int i = blockIdx.x * blockDim.x + threadIdx.x;
  if (i < n) p[i] = v;
}

__global__ void rsqrt_inplace(float* __restrict__ d, int n) {
  int i = blockIdx.x * blockDim.x + threadIdx.x;
  if (i < n) d[i] = rsqrtf(d[i]);
}

// Fused: int64 -> packed int32 edge conversion + degree count.
// Index stream is read-once: non-temporal so it does not thrash L2.
__global__ void convert_deg(const long long* __restrict__ src, const long long* __restrict__ dst,
                            v2i* __restrict__ ep, float* __restrict__ deg, int E) {
  int e = blockIdx.x * blockDim.x + threadIdx.x;
  if (e >= E) return;
  long long s = __builtin_nontemporal_load(src + e);
  long long d = __builtin_nontemporal_load(dst + e);
  v2i p;
  p.x = (int)s;
  p.y = (int)d;
  ep[e] = p;  // regular store: packed indices may stay resident in 192MB L2
  atomicAdd(&deg[(int)d], 1.0f);
}

// Fallback degree kernel (no workspace for packed indices)
__global__ void deg_edges(const long long* __restrict__ dst, float* __restrict__ deg, int E) {
  int e = blockIdx.x * blockDim.x + threadIdx.x;
  if (e < E) atomicAdd(&deg[(int)__builtin_nontemporal_load(dst + e)], 1.0f);
}

// ---------------- layer-1 GEMM: t[N,4] = x[N,256] @ W1[256,4] via f32 WMMA ----------------
// One wave per 16-row tile; K looped in steps of 4 using V_WMMA_F32_16X16X4_F32.
// W1 zero-padded to 16 cols, packed in LDS in the per-lane B pair layout
// (one ds_load_b64 per step). x is streamed non-temporally (read-once).
__global__ __launch_bounds__(256) void gemm1_wmma(const float* __restrict__ x,
                                                  const float* __restrict__ W1,
                                                  float* __restrict__ t, int n) {
  // sB[pair*32 + lid*2 + c] = Wpad[pair*2 + c][lid]  (pair = k/2, c = 0/1)
  __shared__ float sB[(NF / 2) * 32];  // 4096 floats = 16 KB
  int tid = threadIdx.x;
#pragma unroll
  for (int idx = tid; idx < (NF / 2) * 32; idx += 256) {
    int pair = idx >> 5;
    int rem  = idx & 31;
    int l    = rem >> 1;      // padded column 0..15
    int c    = rem & 1;       // which K of the pair
    int kk   = pair * 2 + c;  // K index 0..255
    sB[idx] = (l < 4) ? W1[kk * 4 + l] : 0.0f;
  }
  __syncthreads();

  int lane = tid & 31;
  int wave = tid >> 5;
  int row0 = (blockIdx.x * 8 + wave) * 16;
  if (row0 >= n) return;

  int half = lane >> 4;   // 0: lanes 0-15 (K=k,k+1), 1: lanes 16-31 (K=k+2,k+3)
  int lid  = lane & 15;
  int koff = half * 2;
  const float* xr = x + (size_t)(row0 + lid) * NF + koff;  // 8B-aligned
  const float* bp = sB + half * 32 + lid * 2;              // + k*16 per step

  v8f acc = {};
#pragma unroll 4
  for (int k = 0; k < NF; k += 4) {
    v2f A = __builtin_nontemporal_load((const v2f*)(xr + k));  // global_load_b64 NT
    v2f B = *(const v2f*)(bp + k * 16);                        // ds_load_b64
    acc = __builtin_amdgcn_wmma_f32_16x16x4_f32(false, A, false, B, (short)0, acc,
                                                false, false);
  }

  // C/D layout: VGPR v -> row = row0 + v + 8*half, col = lid (only cols 0..3 real)
  if (lid < 4) {
    int rbase = row0 + half * 8;
#pragma unroll
    for (int v = 0; v < 8; ++v)
      t[(size_t)(rbase + v) * 4 + lid] = acc[v];
  }
}

// ---------------- edge aggregation: agg[dst] += dinv[src]*dinv[dst]*h[src] ----------------
// Packed int32 index variant (h row stride 4; C = live columns).
template <int C>
__global__ void edge_agg_p(const v2i* __restrict__ ep, const float* __restrict__ dinv,
                           const float* __restrict__ h, float* __restrict__ agg, int E) {
  int e = blockIdx.x * blockDim.x + threadIdx.x;
  if (e >= E) return;
  v2i p = ep[e];  // global_load_b64 (hot in L2 after conversion pass)
  int s = p.x, d = p.y;
  float c = dinv[s] * dinv[d];
  if constexpr (C == 4) {
    v4f hv = *(const v4f*)(h + (size_t)s * 4);  // global_load_b128
    atomicAdd(&agg[(size_t)d * 4 + 0], c * hv.x);
    atomicAdd(&agg[(size_t)d * 4 + 1], c * hv.y);
    atomicAdd(&agg[(size_t)d * 4 + 2], c * hv.z);
    atomicAdd(&agg[(size_t)d * 4 + 3], c * hv.w);
  } else {
    v2f hv = *(const v2f*)(h + (size_t)s * 4);  // global_load_b64
    atomicAdd(&agg[(size_t)d * 2 + 0], c * hv.x);
    atomicAdd(&agg[(size_t)d * 2 + 1], c * hv.y);
  }
}

// int64 fallback variant (indices streamed non-temporally)
template <int C>
__global__ void edge_agg64(const long long* __restrict__ src, const long long* __restrict__ dst,
                           const float* __restrict__ dinv, const float* __restrict__ h,
                           float* __restrict__ agg, int E) {
  int e = blockIdx.x * blockDim.x + threadIdx.x;
  if (e >= E) return;
  int s = (int)__builtin_nontemporal_load(src + e);
  int d = (int)__builtin_nontemporal_load(dst + e);
  float c = dinv[s] * dinv[d];
  if constexpr (C == 4) {
    v4f hv = *(const v4f*)(h + (size_t)s * 4);
    atomicAdd(&agg[(size_t)d * 4 + 0], c * hv.x);
    atomicAdd(&agg[(size_t)d * 4 + 1], c * hv.y);
    atomicAdd(&agg[(size_t)d * 4 + 2], c * hv.z);
    atomicAdd(&agg[(size_t)d * 4 + 3], c * hv.w);
  } else {
    v2f hv = *(const v2f*)(h + (size_t)s * 4);
    atomicAdd(&agg[(size_t)d * 2 + 0], c * hv.x);
    atomicAdd(&agg[(size_t)d * 2 + 1], c * hv.y);
  }
}

// ---------------- node update: h = tanh(agg + dinv^2 * t + b); t = h @ Wn ----------------
template <int C, int CN>
__global__ void node_update(const float* __restrict__ agg, const float* __restrict__ dinv,
                            const float* __restrict__ b, const float* __restrict__ Wn,
                            float* __restrict__ t, int n) {
  int i = blockIdx.x * blockDim.x + threadIdx.x;
  if (i >= n) return;
  float di2 = dinv[i] * dinv[i];
  float h[C];
#pragma unroll
  for (int j = 0; j < C; ++j)
    h[j] = tanhf(agg[(size_t)i * C + j] + di2 * t[(size_t)i * 4 + j] + b[j]);
  float o[CN];
#pragma unroll
  for (int j = 0; j < CN; ++j) {
    float a = 0.0f;
#pragma unroll
    for (int k = 0; k < C; ++k) a += h[k] * Wn[k * CN + j];
    o[j] = a;
  }
#pragma unroll
  for (int j = 0; j < CN; ++j) t[(size_t)i * 4 + j] = o[j];  // own-row in-place: safe
}

// ---------------- final layer: h3 = tanh(...), out = h3 @ Wc + bc ----------------
__global__ void node_final(const float* __restrict__ agg, const float* __restrict__ dinv,
                           const float* __restrict__ t, const float* __restrict__ b3,
                           const float* __restrict__ Wc, const float* __restrict__ bc,
                           float* __restrict__ out, int n) {
  int i = blockIdx.x * blockDim.x + threadIdx.x;
  if (i >= n) return;
  float di2 = dinv[i] * dinv[i];
  float h0 = tanhf(agg[(size_t)i * 2 + 0] + di2 * t[(size_t)i * 4 + 0] + b3[0]);
  float h1 = tanhf(agg[(size_t)i * 2 + 1] + di2 * t[(size_t)i * 4 + 1] + b3[1]);
  // outputs are never re-read on device: non-temporal stores
  __builtin_nontemporal_store(h0 * Wc[0] + h1 * Wc[1] + bc[0], out + i);
  __builtin_nontemporal_store(h0, out + (size_t)n + 2 * i + 0);
  __builtin_nontemporal_store(h1, out + (size_t)n + 2 * i + 1);
}

extern "C" void kernel_launch(void* const* d_in, const int* in_sizes, int n_in,
                              void* d_out, int out_size, void* d_ws, size_t ws_size,
                              hipStream_t stream) {
  const float*     x  = (const float*)d_in[0];
  const long long* ei = (const long long*)d_in[1];  // int64 [2, E]
  const float*     W1 = (const float*)d_in[2];
  const float*     b1 = (const float*)d_in[3];
  const float*     W2 = (const float*)d_in[4];
  const float*     b2 = (const float*)d_in[5];
  const float*     W3 = (const float*)d_in[6];
  const float*     b3 = (const float*)d_in[7];
  const float*     Wc = (const float*)d_in[8];
  const float*     bc = (const float*)d_in[9];

  int N = in_sizes[0] / NF;  // 200000
  int E = in_sizes[1] / 2;   // 12800000

  // workspace: dinv[N] | t[N*4] | agg[N*4] | (optional) packed int2 edges[E]
  float* dinv = (float*)d_ws;
  float* t    = dinv + N;
  float* agg  = t + (size_t)4 * N;
  v2i*   ep   = (v2i*)(agg + (size_t)4 * N);

  size_t base_bytes = (size_t)9 * N * sizeof(float);
  bool   useI32     = ws_size >= base_bytes + (size_t)E * sizeof(v2i);

  const long long* srcp = ei;
  const long long* dstp = ei + E;

  dim3 blk(256);
  auto g = [](int n) { return dim3((unsigned)((n + 255) / 256)); };

  // degree (+ optional index compaction), then dinv = rsqrt(deg)
  fill_f32<<<g(N), blk, 0, stream>>>(dinv, 1.0f, N);
  if (useI32)
    convert_deg<<<g(E), blk, 0, stream>>>(srcp, dstp, ep, dinv, E);
  else
    deg_edges<<<g(E), blk, 0, stream>>>(dstp, dinv, E);
  rsqrt_inplace<<<g(N), blk, 0, stream>>>(dinv, N);

  // layer 1: t = x @ W1 (WMMA), aggregate, fuse tanh + (h1 @ W2)
  gemm1_wmma<<<dim3((unsigned)((N + 127) / 128)), 256, 0, stream>>>(x, W1, t, N);
  fill_f32<<<g(4 * N), blk, 0, stream>>>(agg, 0.0f, 4 * N);
  if (useI32)
    edge_agg_p<4><<<g(E), blk, 0, stream>>>(ep, dinv, t, agg, E);
  else
    edge_agg64<4><<<g(E), blk, 0, stream>>>(srcp, dstp, dinv, t, agg, E);
  node_update<4, 4><<<g(N), blk, 0, stream>>>(agg, dinv, b1, W2, t, N);

  // layer 2: aggregate, fuse tanh + (h2 @ W3) -> t cols 0..1 (stride 4)
  fill_f32<<<g(4 * N), blk, 0, stream>>>(agg, 0.0f, 4 * N);
  if (useI32)
    edge_agg_p<4><<<g(E), blk, 0, stream>>>(ep, dinv, t, agg, E);
  else
    edge_agg64<4><<<g(E), blk, 0, stream>>>(srcp, dstp, dinv, t, agg, E);
  node_update<4, 2><<<g(N), blk, 0, stream>>>(agg, dinv, b2, W3, t, N);

  // layer 3: aggregate (C=2), final tanh + classifier head, write both outputs
  fill_f32<<<g(2 * N), blk, 0, stream>>>(agg, 0.0f, 2 * N);
  if (useI32)
    edge_agg_p<2><<<g(E), blk, 0, stream>>>(ep, dinv, t, agg, E);
  else
    edge_agg64<2><<<g(E), blk, 0, stream>>>(srcp, dstp, dinv, t, agg, E);
  node_final<<<g(N), blk, 0, stream>>>(agg, dinv, t, b3, Wc, bc, (float*)d_out, N);
}